// LowRankGNNLayer_103079215401
// MI455X (gfx1250) — compile-verified
//
#include <hip/hip_runtime.h>

typedef float v2f __attribute__((ext_vector_type(2)));
typedef float v8f __attribute__((ext_vector_type(8)));

#define B_ROWS   20000
#define N1_ROWS  80000
#define E_EDGES  800000
#define C_DIM    256
#define NUM_D    32
#define NB       8
#define NUM_M    4096
#define NUM_N    200000
#define NTOT     100000   // B_ROWS + N1_ROWS

// ---------------------------------------------------------------------------
// Kernel 1: build x_input = concat(x, gathered codebook[:, :, :32])
// one float4 per thread; 100000*64 float4 = 6.4M threads
// ---------------------------------------------------------------------------
__global__ __launch_bounds__(256) void k_build_xinput(
    const float* __restrict__ x, const float* __restrict__ codebook,
    const int* __restrict__ c_indices, const int* __restrict__ fo,
    float* __restrict__ x_input)
{
    int gid = blockIdx.x * 256 + threadIdx.x;          // 0 .. 6.4M-1
    int r   = gid >> 6;                                // row
    int c4  = (gid & 63) * 4;                          // col (float4 aligned)
    float4 v;
    if (r < B_ROWS) {
        v = *(const float4*)(x + (size_t)r * C_DIM + c4);
    } else {
        int rr  = r - B_ROWS;
        int nb  = c4 >> 5;                             // which 32-wide chunk
        int d   = c4 & 31;
        int idx = c_indices[(size_t)nb * NUM_N + fo[rr]];
        v = *(const float4*)(codebook + (((size_t)nb * NUM_M + idx) * 2 * NUM_D) + d);
    }
    *(float4*)(x_input + (size_t)r * C_DIM + c4) = v;
}

// ---------------------------------------------------------------------------
// Kernel 2: init x_output rows with b_conv (segment_sum bias) + zero reducer
// ---------------------------------------------------------------------------
__global__ __launch_bounds__(256) void k_init_out(
    const float* __restrict__ b_conv, float* __restrict__ x_output,
    float* __restrict__ wsum)
{
    int gid = blockIdx.x * 256 + threadIdx.x;
    int c4  = (gid & 63) * 4;
    float4 bv = *(const float4*)(b_conv + c4);
    *(float4*)(x_output + (size_t)gid * 4) = bv;       // gid*4 == r*256 + c4
    if (gid == 0) *wsum = 0.0f;
}

// ---------------------------------------------------------------------------
// Kernel 3: xw = x_input @ W_conv  (M=100000, K=N=256), f32 WMMA 16x16x4
// workgroup = 8 waves -> 32x256 output tile; LDS-staged K panels of 32
// ---------------------------------------------------------------------------
#define KCHUNK  32
#define BSTRIDE 260      // padded LDS row stride (floats) for W panel
#define ASTRIDE 36       // padded LDS row stride for A tile

__global__ __launch_bounds__(256) void k_gemm_conv(
    const float* __restrict__ A, const float* __restrict__ W,
    float* __restrict__ Out)
{
    __shared__ float sB[KCHUNK * BSTRIDE];   // 32 x 256 W panel (padded)
    __shared__ float sA[32 * ASTRIDE];       // 32 x 32 A tile   (padded)

    const int tid   = threadIdx.x;
    const int lane  = tid & 31;
    const int wave  = tid >> 5;
    const int mblk  = blockIdx.x * 32;
    const int wrow  = (wave >> 2) * 16;      // 0 / 16
    const int wcol  = (wave & 3) * 64;       // 0,64,128,192
    const int l15   = lane & 15;
    const int khalf = (lane >> 4) * 2;       // 0 / 2 (f32 A/B fragment phase)

    v8f acc0 = {}, acc1 = {}, acc2 = {}, acc3 = {};

    for (int k0 = 0; k0 < C_DIM; k0 += KCHUNK) {
        // stage W panel: 32x256 floats, 256 threads x 8 float4
        #pragma unroll
        for (int i = 0; i < 8; ++i) {
            int e  = tid + i * 256;               // 0..2047 float4 slots
            int kk = e >> 6;                      // 64 float4 per row
            int c4 = (e & 63) * 4;
            float4 v = *(const float4*)(W + (size_t)(k0 + kk) * C_DIM + c4);
            float* d = &sB[kk * BSTRIDE + c4];
            d[0] = v.x; d[1] = v.y; d[2] = v.z; d[3] = v.w;
        }
        // stage A tile: 32x32 floats, 256 threads x 1 float4
        {
            int row = tid >> 3;
            int c4  = (tid & 7) * 4;
            float4 v = *(const float4*)(A + (size_t)(mblk + row) * C_DIM + k0 + c4);
            float* d = &sA[row * ASTRIDE + c4];
            d[0] = v.x; d[1] = v.y; d[2] = v.z; d[3] = v.w;
        }
        __syncthreads();

        #pragma unroll
        for (int kk = 0; kk < KCHUNK; kk += 4) {
            v2f a;
            const float* ap = &sA[(wrow + l15) * ASTRIDE + kk + khalf];
            a[0] = ap[0]; a[1] = ap[1];
            const float* bp = &sB[(kk + khalf) * BSTRIDE + wcol + l15];
            v2f b;
            b[0] = bp[0];      b[1] = bp[BSTRIDE];
            acc0 = __builtin_amdgcn_wmma_f32_16x16x4_f32(false, a, false, b, (short)0, acc0, false, false);
            b[0] = bp[16];     b[1] = bp[BSTRIDE + 16];
            acc1 = __builtin_amdgcn_wmma_f32_16x16x4_f32(false, a, false, b, (short)0, acc1, false, false);
            b[0] = bp[32];     b[1] = bp[BSTRIDE + 32];
            acc2 = __builtin_amdgcn_wmma_f32_16x16x4_f32(false, a, false, b, (short)0, acc2, false, false);
            b[0] = bp[48];     b[1] = bp[BSTRIDE + 48];
            acc3 = __builtin_amdgcn_wmma_f32_16x16x4_f32(false, a, false, b, (short)0, acc3, false, false);
        }
        __syncthreads();
    }

    // C/D layout: VGPR i -> M = i + (lane>=16 ? 8 : 0); N = lane&15
    const int rbase = mblk + wrow + (lane >> 4) * 8;
    #pragma unroll
    for (int i = 0; i < 8; ++i) {
        float* o = Out + (size_t)(rbase + i) * C_DIM + wcol + l15;
        o[0]  = acc0[i];
        o[16] = acc1[i];
        o[32] = acc2[i];
        o[48] = acc3[i];
    }
}

// ---------------------------------------------------------------------------
// Kernel 4: edge scatter. One wave per edge; 8 f32 atomics/lane into L2-resident
// x_output (102 MB < 192 MB L2).
// ---------------------------------------------------------------------------
__global__ __launch_bounds__(256) void k_scatter(
    const int* __restrict__ ei, const float* __restrict__ ew,
    const float* __restrict__ xw, float* __restrict__ xout)
{
    int e    = blockIdx.x * 8 + (threadIdx.x >> 5);
    int lane = threadIdx.x & 31;
    int src  = ei[e];
    int dst  = ei[E_EDGES + e];
    float w  = ew[e];
    const float4* sp = (const float4*)(xw + (size_t)src * C_DIM);
    float* dp = xout + (size_t)dst * C_DIM;
    #pragma unroll
    for (int i = 0; i < 2; ++i) {
        int c4 = lane + i * 32;
        float4 v = sp[c4];
        atomicAdd(dp + c4 * 4 + 0, v.x * w);
        atomicAdd(dp + c4 * 4 + 1, v.y * w);
        atomicAdd(dp + c4 * 4 + 2, v.z * w);
        atomicAdd(dp + c4 * 4 + 3, v.w * w);
    }
}

// ---------------------------------------------------------------------------
// Kernel 5: info_backward partial reduce. One block per first-order row;
// grad gathered directly from codebook[:, :, 32:] (no materialization).
// ---------------------------------------------------------------------------
__global__ __launch_bounds__(256) void k_info(
    const float* __restrict__ xout, const float* __restrict__ codebook,
    const int* __restrict__ c_indices, const int* __restrict__ fo,
    float* __restrict__ wsum)
{
    __shared__ float red[256];
    int r = blockIdx.x;
    int c = threadIdx.x;
    int nb  = c >> 5;
    int d   = c & 31;
    int idx = c_indices[(size_t)nb * NUM_N + fo[r]];
    float g = codebook[((size_t)nb * NUM_M + idx) * 2 * NUM_D + NUM_D + d];
    float v = xout[(size_t)(B_ROWS + r) * C_DIM + c] * g;
    red[c] = v;
    __syncthreads();
    #pragma unroll
    for (int s = 128; s > 0; s >>= 1) {
        if (c < s) red[c] += red[c + s];
        __syncthreads();
    }
    if (c == 0) atomicAdd(wsum, red[0]);
}

// ---------------------------------------------------------------------------
// Kernel 6: out = x_output[:B] @ W_t + x @ W_skip + (b_t + b_skip)
// fused concat-K WMMA GEMM, bias epilogue, writes d_out directly.
// ---------------------------------------------------------------------------
__global__ __launch_bounds__(256) void k_gemm_out(
    const float* __restrict__ A0, const float* __restrict__ A1,
    const float* __restrict__ W0, const float* __restrict__ W1,
    const float* __restrict__ b0, const float* __restrict__ b1,
    float* __restrict__ Out)
{
    __shared__ float sB[KCHUNK * BSTRIDE];
    __shared__ float sA[32 * ASTRIDE];

    const int tid   = threadIdx.x;
    const int lane  = tid & 31;
    const int wave  = tid >> 5;
    const int mblk  = blockIdx.x * 32;
    const int wrow  = (wave >> 2) * 16;
    const int wcol  = (wave & 3) * 64;
    const int l15   = lane & 15;
    const int khalf = (lane >> 4) * 2;

    v8f acc0 = {}, acc1 = {}, acc2 = {}, acc3 = {};

    for (int pass = 0; pass < 2; ++pass) {
        const float* A = pass ? A1 : A0;
        const float* W = pass ? W1 : W0;
        for (int k0 = 0; k0 < C_DIM; k0 += KCHUNK) {
            #pragma unroll
            for (int i = 0; i < 8; ++i) {
                int e  = tid + i * 256;
                int kk = e >> 6;
                int c4 = (e & 63) * 4;
                float4 v = *(const float4*)(W + (size_t)(k0 + kk) * C_DIM + c4);
                float* d = &sB[kk * BSTRIDE + c4];
                d[0] = v.x; d[1] = v.y; d[2] = v.z; d[3] = v.w;
            }
            {
                int row = tid >> 3;
                int c4  = (tid & 7) * 4;
                float4 v = *(const float4*)(A + (size_t)(mblk + row) * C_DIM + k0 + c4);
                float* d = &sA[row * ASTRIDE + c4];
                d[0] = v.x; d[1] = v.y; d[2] = v.z; d[3] = v.w;
            }
            __syncthreads();
            #pragma unroll
            for (int kk = 0; kk < KCHUNK; kk += 4) {
                v2f a;
                const float* ap = &sA[(wrow + l15) * ASTRIDE + kk + khalf];
                a[0] = ap[0]; a[1] = ap[1];
                const float* bp = &sB[(kk + khalf) * BSTRIDE + wcol + l15];
                v2f b;
                b[0] = bp[0];  b[1] = bp[BSTRIDE];
                acc0 = __builtin_amdgcn_wmma_f32_16x16x4_f32(false, a, false, b, (short)0, acc0, false, false);
                b[0] = bp[16]; b[1] = bp[BSTRIDE + 16];
                acc1 = __builtin_amdgcn_wmma_f32_16x16x4_f32(false, a, false, b, (short)0, acc1, false, false);
                b[0] = bp[32]; b[1] = bp[BSTRIDE + 32];
                acc2 = __builtin_amdgcn_wmma_f32_16x16x4_f32(false, a, false, b, (short)0, acc2, false, false);
                b[0] = bp[48]; b[1] = bp[BSTRIDE + 48];
                acc3 = __builtin_amdgcn_wmma_f32_16x16x4_f32(false, a, false, b, (short)0, acc3, false, false);
            }
            __syncthreads();
        }
    }

    const int rbase = mblk + wrow + (lane >> 4) * 8;
    const int col   = wcol + l15;
    float bias0 = b0[col]      + b1[col];
    float bias1 = b0[col + 16] + b1[col + 16];
    float bias2 = b0[col + 32] + b1[col + 32];
    float bias3 = b0[col + 48] + b1[col + 48];
    #pragma unroll
    for (int i = 0; i < 8; ++i) {
        float* o = Out + (size_t)(rbase + i) * C_DIM + col;
        o[0]  = acc0[i] + bias0;
        o[16] = acc1[i] + bias1;
        o[32] = acc2[i] + bias2;
        o[48] = acc3[i] + bias3;
    }
}

// ---------------------------------------------------------------------------
// Kernel 7: finalize scalar: d_out[B*C] = wsum * warm_up_rate
// ---------------------------------------------------------------------------
__global__ void k_final(const float* __restrict__ wsum,
                        const float* __restrict__ warm,
                        float* __restrict__ out_scalar)
{
    *out_scalar = wsum[0] * warm[0];
}

// ---------------------------------------------------------------------------
extern "C" void kernel_launch(void* const* d_in, const int* in_sizes, int n_in,
                              void* d_out, int out_size, void* d_ws, size_t ws_size,
                              hipStream_t stream)
{
    const float* x        = (const float*)d_in[0];
    const float* ew       = (const float*)d_in[1];
    const float* codebook = (const float*)d_in[2];
    const float* Wconv    = (const float*)d_in[3];
    const float* bconv    = (const float*)d_in[4];
    const float* Wt       = (const float*)d_in[5];
    const float* bt       = (const float*)d_in[6];
    const float* Wskip    = (const float*)d_in[7];
    const float* bskip    = (const float*)d_in[8];
    const float* warm     = (const float*)d_in[9];
    const int*   cidx     = (const int*)d_in[10];
    const int*   fo       = (const int*)d_in[11];
    const int*   ei       = (const int*)d_in[12];
    float* out = (float*)d_out;

    float* x_input = (float*)d_ws;
    float* xw      = x_input + (size_t)NTOT * C_DIM;
    float* xout    = xw      + (size_t)NTOT * C_DIM;
    float* wsum    = xout    + (size_t)NTOT * C_DIM;

    k_build_xinput<<<(NTOT * 64) / 256, 256, 0, stream>>>(x, codebook, cidx, fo, x_input);
    k_init_out    <<<(NTOT * 64) / 256, 256, 0, stream>>>(bconv, xout, wsum);
    k_gemm_conv   <<<NTOT / 32, 256, 0, stream>>>(x_input, Wconv, xw);
    k_scatter     <<<E_EDGES / 8, 256, 0, stream>>>(ei, ew, xw, xout);
    k_info        <<<N1_ROWS, 256, 0, stream>>>(xout, codebook, cidx, fo, wsum);
    k_gemm_out    <<<B_ROWS / 32, 256, 0, stream>>>(xout, x, Wt, Wskip, bt, bskip, out);
    k_final       <<<1, 1, 0, stream>>>(wsum, warm, out + (size_t)B_ROWS * C_DIM);
}